// DTCWTLayerQshift_bpInv_30253749633568
// MI455X (gfx1250) — compile-verified
//
#include <hip/hip_runtime.h>

typedef float v2f __attribute__((ext_vector_type(2)));
typedef float v8f __attribute__((ext_vector_type(8)));

#define SCQ 0.70710678118654752440f

// base input-row of the 24-wide band window for output block t (16 rows),
// filter f (0:g0 14-tap, 1:g1 14-tap, 2:g2 18-tap), r = 128. Always even.
__device__ __forceinline__ int base_of(int f, int t) {
    int m2 = (f == 2) ? 9 : 7;
    int b = 8 * t + 1 - m2;
    if (b < 0) b = 0;
    if (b > 104) b = 104;   // 128 - 24
    return b;
}

// One stage-A accumulation pass over the 24-wide band: 6 chained WMMAs.
// SRC: 0=Yl, 1=HL(c2q subbands 2,3), 2=LH(0,5), 3=HH(1,4). F selects filter.
// A = weight chunk (16x4), B = gathered data chunk (4x16), all compile-time paths.
template<int SRC, int F>
__device__ __forceinline__ v8f stageA_accum(v8f acc,
                                            const float* __restrict__ Yl,
                                            const float* __restrict__ Yhr,
                                            const float* __restrict__ Yhi,
                                            const float (* __restrict__ sWt)[25],
                                            int img, int t, int hbase,
                                            int m, int half, bool heven) {
    const int base = base_of(F, t);
    const int h = hbase + m;
#pragma unroll
    for (int ks = 0; ks < 6; ++ks) {
        const int kk = 4 * ks + (half << 1);
        const int w = base + kk;                 // even
        v2f a, b;
        a.x = sWt[m][kk];
        a.y = sWt[m][kk + 1];
        if (SRC == 0) {
            b = *(const v2f*)(Yl + ((size_t)img * 128 + h) * 128 + w);
        } else {
            constexpr int s0 = (SRC == 1) ? 2 : (SRC == 2) ? 0 : 1;
            constexpr int s1 = (SRC == 1) ? 3 : (SRC == 2) ? 5 : 4;
            const int hi = h >> 1, wi = w >> 1;  // (w, w+1) share one cell
            const size_t i0 = (((size_t)img * 6 + s0) * 64 + hi) * 64 + wi;
            const size_t i1 = (((size_t)img * 6 + s1) * 64 + hi) * 64 + wi;
            const float r0 = Yhr[i0], r1 = Yhr[i1];
            const float q0 = Yhi[i0], q1 = Yhi[i1];
            b.x = SCQ * (heven ? (r0 + r1) : (q0 - q1));   // x1 | x3
            b.y = SCQ * (heven ? (q0 + q1) : (r1 - r0));   // x2 | x4
        }
        acc = __builtin_amdgcn_wmma_f32_16x16x4_f32(
                false, a, false, b, (short)0, acc, false, false);
    }
    return acc;
}

__global__ __launch_bounds__(256)
void dtcwt_inv_fused(const float* __restrict__ Yl,
                     const float* __restrict__ Yhr,
                     const float* __restrict__ Yhi,
                     const float* __restrict__ g0a, const float* __restrict__ g0b,
                     const float* __restrict__ g1a, const float* __restrict__ g1b,
                     const float* __restrict__ g2a, const float* __restrict__ g2b,
                     float* __restrict__ out) {
    // Intermediates y1^T,y2^T,y2bp^T for this (image, w-quarter): [w_local 64][h 128], pitch 132.
    __shared__ float sYT[3][64][132];
    // Banded polyphase weight matrices W[filter][out-block t][16 out rows][24 band cols], pitch 25.
    __shared__ float sW[3][16][16][25];
    // Per-wave 16x16 transpose staging for coalesced output stores.
    __shared__ float sTr[8][16][17];

    const int img     = blockIdx.x;   // b*64+ch, 0..511
    const int quarter = blockIdx.y;   // W quarter, 0..3
    const int tid  = threadIdx.x;
    const int wave = tid >> 5;
    const int lane = tid & 31;
    const int m    = lane & 15;       // M (A) / N (B,D) index within tile
    const int half = lane >> 4;       // K-pair select within a 4-wide chunk
    const bool heven = ((m & 1) == 0);

    // Warm L2/prefetch path for this image's low-pass input.
    __builtin_prefetch(Yl + ((size_t)img * 128 + (tid & 127)) * 128, 0, 1);

    // ---- Build the 48 weight matrices (one thread each) -------------------
    if (tid < 48) {
        const int f = tid >> 4, t = tid & 15;
        const float* ha = (f == 0) ? g0b : ((f == 1) ? g1b : g2b); // colifilt(X, ha=*b, hb=*a)
        const float* hb = (f == 0) ? g0a : ((f == 1) ? g1a : g2a);
        const int  m2   = (f == 2) ? 9 : 7;                        // taps per phase (m2 odd branch)
        const bool high = (f != 0);                                // g0 lowpass, g1/g2 highpass
        const int  base = base_of(f, t);
        float* Wp = &sW[f][t][0][0];
        for (int x = 0; x < 16 * 25; ++x) Wp[x] = 0.0f;
        for (int p = 0; p < 4; ++p) {
            // phase p uses selA (off=1) / selB (off=2); swapped for highpass
            const int off = ((p & 1) == 0) ? (high ? 2 : 1) : (high ? 1 : 2);
            for (int k = 0; k < m2; ++k) {
                float tap;
                if (p == 0)      tap = ha[2 * k + 1];  // hao
                else if (p == 1) tap = hb[2 * k + 1];  // hbo
                else if (p == 2) tap = ha[2 * k];      // hae
                else             tap = hb[2 * k];      // hbe
                for (int i = 0; i < 4; ++i) {
                    int q = 2 * (4 * t + i + k) + off - m2;  // padded row index
                    if (q < 0)    q = -q - 1;                // symmetric reflect
                    if (q >= 128) q = 255 - q;
                    const int col = q - base;
                    if (col >= 0 && col < 24) sW[f][t][4 * i + p][col] += tap;
                }
            }
        }
    }
    __syncthreads();

    // ---- Stage A: row-direction upsampling filter (along W) ---------------
    // Out^T(16 wout x 16 h) = W(16x24) @ X^T(24x16); A=weights, B=gathered data.
    // arr 0: y1 = Yl*g0(low) + HL*g1(high); arr 1: y2 = LH*g0; arr 2: y2bp = HH*g2.
    for (int job = wave; job < 32; job += 8) {
        const int tb = job >> 3, hb = job & 7;
        const int t = quarter * 4 + tb, hbase = hb * 16;
        v8f acc = {};
        acc = stageA_accum<0, 0>(acc, Yl, Yhr, Yhi, sW[0][t], img, t, hbase, m, half, heven);
        acc = stageA_accum<1, 1>(acc, Yl, Yhr, Yhi, sW[1][t], img, t, hbase, m, half, heven);
#pragma unroll
        for (int v = 0; v < 8; ++v)
            sYT[0][tb * 16 + v + half * 8][hbase + m] = acc[v];
    }
    for (int job = wave; job < 32; job += 8) {
        const int tb = job >> 3, hb = job & 7;
        const int t = quarter * 4 + tb, hbase = hb * 16;
        v8f acc = {};
        acc = stageA_accum<2, 0>(acc, Yl, Yhr, Yhi, sW[0][t], img, t, hbase, m, half, heven);
#pragma unroll
        for (int v = 0; v < 8; ++v)
            sYT[1][tb * 16 + v + half * 8][hbase + m] = acc[v];
    }
    for (int job = wave; job < 32; job += 8) {
        const int tb = job >> 3, hb = job & 7;
        const int t = quarter * 4 + tb, hbase = hb * 16;
        v8f acc = {};
        acc = stageA_accum<3, 2>(acc, Yl, Yhr, Yhi, sW[2][t], img, t, hbase, m, half, heven);
#pragma unroll
        for (int v = 0; v < 8; ++v)
            sYT[2][tb * 16 + v + half * 8][hbase + m] = acc[v];
    }
    __syncthreads();

    // ---- Stage B: column-direction upsampling filter (along H) + sum ------
    // Out^T(16 w x 16 hout) = Y^T(16x24) @ W^T(24x16); A=LDS data, B=weights.
    for (int job = wave; job < 64; job += 8) {
        const int t  = job >> 2;          // hout block 0..15
        const int wb = job & 3;           // local w block 0..3
        v8f acc = {};
#pragma unroll
        for (int arr = 0; arr < 3; ++arr) {
            const int f    = arr;         // y1:g0 low, y2:g1 high, y2bp:g2 high
            const int base = base_of(f, t);
#pragma unroll
            for (int ks = 0; ks < 6; ++ks) {
                const int kk = 4 * ks + (half << 1);
                v2f a, b;
                a.x = sYT[arr][wb * 16 + m][base + kk];
                a.y = sYT[arr][wb * 16 + m][base + kk + 1];
                b.x = sW[f][t][m][kk];
                b.y = sW[f][t][m][kk + 1];
                acc = __builtin_amdgcn_wmma_f32_16x16x4_f32(
                        false, a, false, b, (short)0, acc, false, false);
            }
        }
        // D is (M=w_local, N=hout_local): transpose via per-wave LDS, store coalesced.
#pragma unroll
        for (int v = 0; v < 8; ++v)
            sTr[wave][v + half * 8][m] = acc[v];
        asm volatile("s_wait_dscnt 0" ::: "memory");
        const int wglobal = quarter * 64 + wb * 16;
#pragma unroll
        for (int j = 0; j < 8; ++j) {
            const int hr  = 2 * j + half;
            const float v = sTr[wave][m][hr];
            out[((size_t)img * 256 + (t * 16 + hr)) * 256 + wglobal + m] = v;
        }
    }
}

extern "C" void kernel_launch(void* const* d_in, const int* in_sizes, int n_in,
                              void* d_out, int out_size, void* d_ws, size_t ws_size,
                              hipStream_t stream) {
    (void)in_sizes; (void)n_in; (void)out_size; (void)d_ws; (void)ws_size;
    const float* Yl  = (const float*)d_in[0];
    const float* Yhr = (const float*)d_in[1];
    const float* Yhi = (const float*)d_in[2];
    const float* g0a = (const float*)d_in[3];
    const float* g0b = (const float*)d_in[4];
    const float* g1a = (const float*)d_in[5];
    const float* g1b = (const float*)d_in[6];
    const float* g2a = (const float*)d_in[7];
    const float* g2b = (const float*)d_in[8];
    float* out = (float*)d_out;

    dim3 grid(512, 4);   // (b*ch images, W quarters)
    dtcwt_inv_fused<<<grid, 256, 0, stream>>>(Yl, Yhr, Yhi,
                                              g0a, g0b, g1a, g1b, g2a, g2b, out);
}